// GraphAttention_42314017800356
// MI455X (gfx1250) — compile-verified
//
#include <hip/hip_runtime.h>
#include <math.h>

typedef __attribute__((ext_vector_type(2))) float v2f;
typedef __attribute__((ext_vector_type(4))) float v4f;
typedef __attribute__((ext_vector_type(8))) float v8f;

#define ALPHA 0.2f
#define DIM    67
#define KPAD   68
#define NSAMP  32
#define DOUT   64
#define WAVES_PER_BLOCK 4
#define BLOCK  (WAVES_PER_BLOCK * 32)

#define GSTRIDE (NSAMP * DIM)        // 2144 floats per (b,p) tile
#define GSLOT   (GSTRIDE + 4)        // +4 slack floats (zeroed), keeps 16B stride
#define BROW    160                  // interleaved-a row stride (floats): bank-stagger rows
#define BROWS   (KPAD / 2)           // 34 K-pair rows

__global__ __launch_bounds__(BLOCK) void ga_wmma_kernel(
    const float* __restrict__ centered,
    const float* __restrict__ grouped,
    const float* __restrict__ amat,
    float* __restrict__ out)
{
    // a, K-pair interleaved: element (k,d) at (k/2)*BROW + 2d + (k&1); K=67 row zeroed.
    __shared__ __align__(16) float s_b[BROWS * BROW];
    // grouped tile, unpadded row stride 67; 4 slack floats zeroed (guards K=67 reads).
    __shared__ __align__(16) float s_g[WAVES_PER_BLOCK][GSLOT];
    __shared__ float s_c[WAVES_PER_BLOCK][KPAD];   // centered vector, [67] = 0

    const int tid  = threadIdx.x;
    const int wid  = tid >> 5;
    const int lane = tid & 31;
    const int half = lane >> 4;      // 0: lanes 0-15, 1: lanes 16-31
    const int l16  = lane & 15;

    const size_t pair = (size_t)blockIdx.x * WAVES_PER_BLOCK + wid;   // b*4096 + p

    // ---- stage a interleaved (block-wide) ----
    for (int idx = tid; idx < KPAD * DOUT; idx += BLOCK) {
        const int k = idx >> 6;
        const int d = idx & 63;
        s_b[(k >> 1) * BROW + d * 2 + (k & 1)] = (k < DIM) ? amat[k * DOUT + d] : 0.0f;
    }
    // ---- stage centered point (per wave) ----
    {
        const float* cp = centered + pair * DIM;
        for (int c = lane; c < KPAD; c += 32)
            s_c[wid][c] = (c < DIM) ? cp[c] : 0.0f;
    }
    // ---- stage grouped tile: b128 global loads -> b128 LDS stores ----
    {
        const v4f* gp4 = (const v4f*)(grouped + pair * (size_t)GSTRIDE);
        v4f* sg4 = (v4f*)&s_g[wid][0];
        for (int i = lane; i < GSTRIDE / 4; i += 32)
            sg4[i] = __builtin_nontemporal_load(gp4 + i);
        if (lane == 0) {   // zero slack so K=67 garbage is finite (a-row 67 = 0 kills it)
            s_g[wid][GSTRIDE + 0] = 0.0f; s_g[wid][GSTRIDE + 1] = 0.0f;
            s_g[wid][GSTRIDE + 2] = 0.0f; s_g[wid][GSTRIDE + 3] = 0.0f;
        }
    }
    __syncthreads();

    const float* sg = s_g[wid];
    const float* sc = s_c[wid];

    // ---- scores = (grouped - centered) @ a via V_WMMA_F32_16X16X4_F32, fully unrolled ----
    v8f acc[2][4] = {};
    #pragma unroll
    for (int k = 0; k < KPAD; k += 4) {
        const int ka = k + half * 2;     // VGPR0 carries K=k (lo lanes) / k+2 (hi); VGPR1: +1
        const float c0 = sc[ka];
        const float c1 = sc[ka + 1];
        v2f af0, af1;
        af0.x = sg[l16 * DIM + ka]            - c0;
        af0.y = sg[l16 * DIM + ka + 1]        - c1;
        af1.x = sg[(16 + l16) * DIM + ka]     - c0;
        af1.y = sg[(16 + l16) * DIM + ka + 1] - c1;
        const float* brow = s_b + (ka >> 1) * BROW;   // one b64 per B fragment
        #pragma unroll
        for (int nt = 0; nt < 4; ++nt) {
            const v2f bf = *(const v2f*)(brow + (nt * 16 + l16) * 2);
            acc[0][nt] = __builtin_amdgcn_wmma_f32_16x16x4_f32(
                false, af0, false, bf, (short)0, acc[0][nt], false, false);
            acc[1][nt] = __builtin_amdgcn_wmma_f32_16x16x4_f32(
                false, af1, false, bf, (short)0, acc[1][nt], false, false);
        }
    }

    // ---- leaky-ReLU, softmax over the 32 samples (rows), weighted feature sum ----
    // C/D layout: VGPR i, lanes 0-15 -> M=i, lanes 16-31 -> M=8+i; column = lane%16 in tile.
    float res[4];
    #pragma unroll
    for (int nt = 0; nt < 4; ++nt) {
        float mx = -INFINITY;
        #pragma unroll
        for (int i = 0; i < 8; ++i) {
            float x0 = acc[0][nt][i]; x0 = (x0 >= 0.0f) ? x0 : ALPHA * x0; acc[0][nt][i] = x0;
            float x1 = acc[1][nt][i]; x1 = (x1 >= 0.0f) ? x1 : ALPHA * x1; acc[1][nt][i] = x1;
            mx = fmaxf(mx, fmaxf(x0, x1));
        }
        mx = fmaxf(mx, __shfl_xor(mx, 16, 32));      // lanes n <-> n+16: full 32-row max

        float se = 0.0f;
        #pragma unroll
        for (int i = 0; i < 8; ++i) {
            const float e0 = __expf(acc[0][nt][i] - mx);
            const float e1 = __expf(acc[1][nt][i] - mx);
            acc[0][nt][i] = e0;
            acc[1][nt][i] = e1;
            se += e0 + e1;
        }
        se += __shfl_xor(se, 16, 32);                // full 32-row sum of exps
        const float inv = 1.0f / se;

        const int col = nt * 16 + l16;               // output channel d in [0,64)
        const float* fcol = sg + 3 + col;            // feature = grouped[s][3 + d]
        float partial = 0.0f;
        #pragma unroll
        for (int i = 0; i < 8; ++i) {
            const int r0 = i + half * 8;             // rows of this lane, M-tile 0
            const int r1 = 16 + i + half * 8;        // rows of this lane, M-tile 1
            partial += acc[0][nt][i] * fcol[r0 * DIM];
            partial += acc[1][nt][i] * fcol[r1 * DIM];
        }
        partial *= inv;
        partial += __shfl_xor(partial, 16, 32);      // combine the two 16-row halves
        res[nt] = partial;
    }

    if (half == 0) {
        float* op = out + pair * DOUT;
        #pragma unroll
        for (int nt = 0; nt < 4; ++nt)
            op[nt * 16 + l16] = res[nt];
    }
}

extern "C" void kernel_launch(void* const* d_in, const int* in_sizes, int n_in,
                              void* d_out, int out_size, void* d_ws, size_t ws_size,
                              hipStream_t stream) {
    const float* centered = (const float*)d_in[0];  // (16,4096,67) f32
    const float* grouped  = (const float*)d_in[1];  // (16,4096,32,67) f32
    const float* amat     = (const float*)d_in[2];  // (67,64) f32
    float* out = (float*)d_out;                     // (16,4096,64) f32

    const int npairs = in_sizes[0] / DIM;           // 16*4096 = 65536
    const int grid   = npairs / WAVES_PER_BLOCK;    // 16384
    ga_wmma_kernel<<<grid, BLOCK, 0, stream>>>(centered, grouped, amat, out);
}